// GCN_42923903156343
// MI455X (gfx1250) — compile-verified
//
#include <hip/hip_runtime.h>
#include <hip/hip_bf16.h>

typedef __attribute__((ext_vector_type(2))) float v2f;
typedef __attribute__((ext_vector_type(8))) float v8f;

#define LDS_STRIDE 132   // 128 + 4 pad: conflict-free ds_load_b64 A-fragment reads
#define TB 256

// ---------------------------------------------------------------------------
// Degree / normalization
// ---------------------------------------------------------------------------
__global__ void deg_init_kernel(float* deg, int n) {
  int i = blockIdx.x * blockDim.x + threadIdx.x;
  if (i < n) deg[i] = 1.0f;                       // self-loop weight (fill_value=1)
}

__global__ void deg_edges_kernel(const int* __restrict__ dst,
                                 const float* __restrict__ w,
                                 float* deg, int ne) {
  int e = blockIdx.x * blockDim.x + threadIdx.x;
  if (e < ne) atomicAdd(&deg[dst[e]], w[e]);
}

__global__ void make_dis_kernel(float* deg, int n) {
  int i = blockIdx.x * blockDim.x + threadIdx.x;
  if (i < n) {
    float d = deg[i];
    deg[i] = (d > 0.0f) ? rsqrtf(d) : 0.0f;       // deg^{-1/2}, inf -> 0
  }
}

// ---------------------------------------------------------------------------
// Dense GEMM  C[M x 128] = A[M x 128] @ W[128 x 128]  via V_WMMA_F32_16X16X4_F32
// Block = 8 waves; wave w owns a 16-col slice (B frags live in 64 VGPRs,
// reused across a grid-stride loop over 16-row tiles). A tile staged in LDS.
// ---------------------------------------------------------------------------
__global__ __launch_bounds__(TB) void gemm_wmma_kernel(
    const float* __restrict__ A, const float* __restrict__ W,
    float* __restrict__ C, int M) {
  __shared__ float As[16 * LDS_STRIDE];
  const int lane = threadIdx.x & 31;
  const int wave = threadIdx.x >> 5;              // 0..7
  const int half = lane >> 4;                     // lane group (K split for A/B)
  const int l16  = lane & 15;
  const int c0   = wave * 16;

  // Preload B fragments: wave's 128x16 slice of W, K-major pairs per lane half.
  v2f bfrag[32];
#pragma unroll
  for (int kk = 0; kk < 32; ++kk) {
    const int k0 = kk * 4 + 2 * half;
    bfrag[kk].x = W[k0 * 128 + c0 + l16];
    bfrag[kk].y = W[(k0 + 1) * 128 + c0 + l16];
  }

  const int mtiles = M >> 4;
  for (int rt = blockIdx.x; rt < mtiles; rt += gridDim.x) {
    // Cooperative load of 16x128 A tile (512 float4 across 256 threads).
    const float* Atile = A + (size_t)rt * 16 * 128;
#pragma unroll
    for (int it = 0; it < 2; ++it) {
      const int idx = threadIdx.x + it * TB;      // 0..511
      const int r  = idx >> 5;
      const int cq = (idx & 31) * 4;
      *(float4*)&As[r * LDS_STRIDE + cq] = *(const float4*)(Atile + r * 128 + cq);
    }
    __syncthreads();

    v8f acc = {};
#pragma unroll
    for (int kk = 0; kk < 32; ++kk) {
      const int k0 = kk * 4 + 2 * half;
      v2f a = *(const v2f*)&As[l16 * LDS_STRIDE + k0];   // 8B-aligned ds_load_b64
      acc = __builtin_amdgcn_wmma_f32_16x16x4_f32(
          /*neg_a=*/false, a, /*neg_b=*/false, bfrag[kk],
          /*c_mod=*/(short)0, acc, /*reuse_a=*/false, /*reuse_b=*/false);
    }

    float* Ctile = C + (size_t)rt * 16 * 128 + c0;
#pragma unroll
    for (int j = 0; j < 8; ++j) {
      const int row = j + half * 8;               // C/D layout: VGPR j -> M=j / j+8
      Ctile[row * 128 + l16] = acc[j];
    }
    __syncthreads();
  }
}

// Scalar tail for M not a multiple of 16 (N=50000 is, but stay safe).
__global__ void gemm_tail_kernel(const float* __restrict__ A,
                                 const float* __restrict__ W,
                                 float* __restrict__ C, int M, int m0) {
  const int idx = blockIdx.x * blockDim.x + threadIdx.x;
  const int rows = M - m0;
  if (idx >= rows * 128) return;
  const int r = m0 + idx / 128, c = idx & 127;
  float s = 0.f;
  for (int k = 0; k < 128; ++k) s += A[(size_t)r * 128 + k] * W[k * 128 + c];
  C[(size_t)r * 128 + c] = s;
}

// ---------------------------------------------------------------------------
// agg[i][:] = bias + dis[i]^2 * hW[i][:]   (self-loop term, norm = dis_i * dis_i)
// ---------------------------------------------------------------------------
__global__ void agg_init_kernel(const float* __restrict__ hW,
                                const float* __restrict__ dis,
                                const float* __restrict__ bias,
                                float* __restrict__ H, int n) {
  const int t = blockIdx.x * blockDim.x + threadIdx.x;
  if (t >= n * 32) return;
  const int i = t >> 5;
  const int q = (t & 31) * 4;
  const float di = dis[i];
  const float nn = di * di;
  const float4 v = *(const float4*)(hW + (size_t)i * 128 + q);
  float4 o;
  o.x = bias[q + 0] + nn * v.x;
  o.y = bias[q + 1] + nn * v.y;
  o.z = bias[q + 2] + nn * v.z;
  o.w = bias[q + 3] + nn * v.w;
  *(float4*)(H + (size_t)i * 128 + q) = o;
}

// ---------------------------------------------------------------------------
// Edge scatter: H[dst] += dis[src]*w*dis[dst] * hW[src]. One wave per edge,
// 4 floats per lane, fp32 global atomics (bandwidth/atomic bound phase).
// ---------------------------------------------------------------------------
__global__ void scatter_edges_kernel(const int* __restrict__ src,
                                     const int* __restrict__ dst,
                                     const float* __restrict__ w,
                                     const float* __restrict__ dis,
                                     const float* __restrict__ hW,
                                     float* __restrict__ H, int ne) {
  const int t = blockIdx.x * blockDim.x + threadIdx.x;
  const int e = t >> 5;
  if (e >= ne) return;
  const int q = (t & 31) * 4;
  const int s = src[e], d = dst[e];
  const float nrm = dis[s] * w[e] * dis[d];
  const float4 v = *(const float4*)(hW + (size_t)s * 128 + q);
  float* out = H + (size_t)d * 128 + q;
  atomicAdd(out + 0, nrm * v.x);
  atomicAdd(out + 1, nrm * v.y);
  atomicAdd(out + 2, nrm * v.z);
  atomicAdd(out + 3, nrm * v.w);
}

// ---------------------------------------------------------------------------
// Fused activation/BatchNorm. mode 0: bn(relu(h)); 1: relu(bn(h)); 2: bn(h).
// ---------------------------------------------------------------------------
__global__ void bn_act_kernel(float* __restrict__ H,
                              const float* __restrict__ mean,
                              const float* __restrict__ var,
                              const float* __restrict__ gamma,
                              const float* __restrict__ beta,
                              int n, int mode) {
  const int idx = blockIdx.x * blockDim.x + threadIdx.x;
  if (idx >= n * 128) return;
  const int f = idx & 127;
  float v = H[idx];
  if (mode == 0) v = fmaxf(v, 0.0f);
  v = (v - mean[f]) * rsqrtf(var[f] + 1e-5f) * gamma[f] + beta[f];
  if (mode == 1) v = fmaxf(v, 0.0f);
  H[idx] = v;
}

// ---------------------------------------------------------------------------
// Global add pool
// ---------------------------------------------------------------------------
__global__ void pool_zero_kernel(float* g, int sz) {
  const int i = blockIdx.x * blockDim.x + threadIdx.x;
  if (i < sz) g[i] = 0.0f;
}

__global__ void pool_add_kernel(const float* __restrict__ H,
                                const int* __restrict__ batch,
                                float* __restrict__ g, int n) {
  const int t = blockIdx.x * blockDim.x + threadIdx.x;
  if (t >= n * 32) return;
  const int i = t >> 5;
  const int q = (t & 31) * 4;
  const int b = batch[i];
  const float4 v = *(const float4*)(H + (size_t)i * 128 + q);
  float* out = g + (size_t)b * 128 + q;
  atomicAdd(out + 0, v.x);
  atomicAdd(out + 1, v.y);
  atomicAdd(out + 2, v.z);
  atomicAdd(out + 3, v.w);
}

// ---------------------------------------------------------------------------
// fc1 (WMMA): Gout = relu(relu(Gin) @ W + b), rows <= 64, single block.
// ---------------------------------------------------------------------------
__global__ __launch_bounds__(TB) void fc1_wmma_kernel(
    const float* __restrict__ Gin, const float* __restrict__ W,
    const float* __restrict__ bias, float* __restrict__ Gout, int rows) {
  __shared__ float As[64 * LDS_STRIDE];
  const int lane = threadIdx.x & 31;
  const int wave = threadIdx.x >> 5;
  const int half = lane >> 4;
  const int l16  = lane & 15;
  const int c0   = wave * 16;

  for (int idx = threadIdx.x; idx < rows * 32; idx += TB) {
    const int r = idx >> 5, cq = (idx & 31) * 4;
    const float4 v = *(const float4*)(Gin + r * 128 + cq);
    As[r * LDS_STRIDE + cq + 0] = fmaxf(v.x, 0.0f);   // relu on pooled features
    As[r * LDS_STRIDE + cq + 1] = fmaxf(v.y, 0.0f);
    As[r * LDS_STRIDE + cq + 2] = fmaxf(v.z, 0.0f);
    As[r * LDS_STRIDE + cq + 3] = fmaxf(v.w, 0.0f);
  }
  __syncthreads();

  v2f bfrag[32];
#pragma unroll
  for (int kk = 0; kk < 32; ++kk) {
    const int k0 = kk * 4 + 2 * half;
    bfrag[kk].x = W[k0 * 128 + c0 + l16];
    bfrag[kk].y = W[(k0 + 1) * 128 + c0 + l16];
  }

  const float bcol = bias[c0 + l16];
  for (int rt = 0; rt < (rows >> 4); ++rt) {
    v8f acc = {};
#pragma unroll
    for (int kk = 0; kk < 32; ++kk) {
      const int k0 = kk * 4 + 2 * half;
      v2f a = *(const v2f*)&As[(rt * 16 + l16) * LDS_STRIDE + k0];
      acc = __builtin_amdgcn_wmma_f32_16x16x4_f32(false, a, false, bfrag[kk],
                                                  (short)0, acc, false, false);
    }
#pragma unroll
    for (int j = 0; j < 8; ++j) {
      const int row = rt * 16 + j + half * 8;
      Gout[row * 128 + c0 + l16] = fmaxf(acc[j] + bcol, 0.0f);
    }
  }
}

// ---------------------------------------------------------------------------
// fc2 (tiny): out[g][o] = sum_k g2[g][k] * W[k][o] + b[o]
// ---------------------------------------------------------------------------
__global__ void fc2_kernel(const float* __restrict__ g2,
                           const float* __restrict__ W,
                           const float* __restrict__ b,
                           float* __restrict__ out, int G) {
  const int idx = blockIdx.x * blockDim.x + threadIdx.x;
  if (idx >= G * 20) return;
  const int gi = idx / 20, o = idx % 20;
  float s = b[o];
  for (int k = 0; k < 128; ++k) s += g2[gi * 128 + k] * W[k * 20 + o];
  out[idx] = s;
}

// ---------------------------------------------------------------------------
extern "C" void kernel_launch(void* const* d_in, const int* in_sizes, int n_in,
                              void* d_out, int out_size, void* d_ws, size_t ws_size,
                              hipStream_t stream) {
  const float* x     = (const float*)d_in[0];
  const int*   ei    = (const int*)d_in[1];     // [2,E] flat (int32: JAX x64 off)
  const float* ew    = (const float*)d_in[2];
  const int*   batch = (const int*)d_in[3];
  const float* Ws    = (const float*)d_in[4];   // [5,128,128]
  const float* bs    = (const float*)d_in[5];   // [5,128]
  const float* gam   = (const float*)d_in[6];
  const float* bet   = (const float*)d_in[7];
  const float* mu    = (const float*)d_in[8];
  const float* var   = (const float*)d_in[9];
  const float* fc1w  = (const float*)d_in[10];
  const float* fc1b  = (const float*)d_in[11];
  const float* fc2w  = (const float*)d_in[12];
  const float* fc2b  = (const float*)d_in[13];
  float* out = (float*)d_out;

  const int n = in_sizes[0] / 128;
  const int E = in_sizes[2];
  const int G = out_size / 20;

  // Workspace layout (floats): hW[n*128] | H[n*128] | dis[n(pad4)] | g[G*128] | g2[G*128]
  float* ws  = (float*)d_ws;
  float* hW  = ws;
  float* Hb  = hW + (size_t)n * 128;
  float* dis = Hb + (size_t)n * 128;
  float* gb  = dis + (((size_t)n + 3) & ~(size_t)3);
  float* g2  = gb + (size_t)G * 128;

  const int*   src = ei;
  const int*   dst = ei + E;

  deg_init_kernel<<<(n + TB - 1) / TB, TB, 0, stream>>>(dis, n);
  deg_edges_kernel<<<(E + TB - 1) / TB, TB, 0, stream>>>(dst, ew, dis, E);
  make_dis_kernel<<<(n + TB - 1) / TB, TB, 0, stream>>>(dis, n);

  const float* in = x;
  for (int i = 0; i < 5; ++i) {
    const float* Wl = Ws + (size_t)i * 128 * 128;
    const int mtiles = n >> 4;
    const int grid = mtiles < 1536 ? (mtiles > 0 ? mtiles : 1) : 1536;
    if (mtiles > 0)
      gemm_wmma_kernel<<<grid, TB, 0, stream>>>(in, Wl, hW, n);
    const int rem = n & 15;
    if (rem)
      gemm_tail_kernel<<<(rem * 128 + TB - 1) / TB, TB, 0, stream>>>(in, Wl, hW, n, n - rem);

    agg_init_kernel<<<(n * 32 + TB - 1) / TB, TB, 0, stream>>>(hW, dis, bs + i * 128, Hb, n);
    scatter_edges_kernel<<<((size_t)E * 32 + TB - 1) / TB, TB, 0, stream>>>(
        src, dst, ew, dis, hW, Hb, E);
    const int mode = (i < 3) ? 0 : (i == 3 ? 1 : 2);
    bn_act_kernel<<<((size_t)n * 128 + TB - 1) / TB, TB, 0, stream>>>(
        Hb, mu + i * 128, var + i * 128, gam + i * 128, bet + i * 128, n, mode);
    in = Hb;
  }

  pool_zero_kernel<<<(G * 128 + TB - 1) / TB, TB, 0, stream>>>(gb, G * 128);
  pool_add_kernel<<<(n * 32 + TB - 1) / TB, TB, 0, stream>>>(Hb, batch, gb, n);
  fc1_wmma_kernel<<<1, TB, 0, stream>>>(gb, fc1w, fc1b, g2, G);
  fc2_kernel<<<(G * 20 + TB - 1) / TB, TB, 0, stream>>>(g2, fc2w, fc2b, out, G);
}